// DFTD3_16123307229504
// MI455X (gfx1250) — compile-verified
//
#include <hip/hip_runtime.h>
#include <math.h>

// ---------------- constants (folded from the reference) ----------------
#define CANG2BOHR 1.8897261254578281f   /* 1/0.529177210544 */
#define CH2EV     27.211386245981f
#define CPREF     (-0.5f * 27.211386245981f)
#define CA1 0.3981f
#define CA2 4.4211f
#define CS8 1.9889f
#define CS6 1.0f
#define CK1 16.0f
#define CK3 (-4.0f)
#define RREF 5
#define SPLIT 4      /* lanes cooperating on one atom in the heavy passes */

typedef __attribute__((ext_vector_type(2))) float v2f;
typedef __attribute__((ext_vector_type(8))) float v8f;

__device__ __forceinline__ float frcp(float x) { return __builtin_amdgcn_rcpf(x); }

__device__ __forceinline__ void atomic_add_f32(float* p, float v) {
  // relaxed agent-scope fadd -> global_atomic_add_f32 on gfx1250
  __hip_atomic_fetch_add(p, v, __ATOMIC_RELAXED, __HIP_MEMORY_SCOPE_AGENT);
}

// Reduce across the SPLIT adjacent lanes that share one atom (segmented wave32).
__device__ __forceinline__ float group_reduce4(float v) {
  v += __shfl_down(v, 2, 4);
  v += __shfl_down(v, 1, 4);
  return v;
}

// Wave32 total via V_WMMA_F32_16X16X4_F32: each lane's partial occupies one
// distinct (m,k) slot of A (the other slot is zero); B = ones(4x16) so
// D[m][n] = rowsum_m. Per-lane sum of the 8 D accumulators yields
// sum(rows 0..7) on the low half-wave and sum(rows 8..15) on the high half;
// one cross-half shuffle finishes the grand total (valid on lane 0).
// Requires EXEC == all ones (guaranteed: no early returns in callers).
__device__ __forceinline__ float wave_sum_wmma(float v) {
  v2f A; A.x = v;    A.y = 0.0f;
  v2f B; B.x = 1.0f; B.y = 1.0f;
  v8f C = {};
  v8f D = __builtin_amdgcn_wmma_f32_16x16x4_f32(
      /*neg_a=*/false, A, /*neg_b=*/false, B,
      /*c_mod=*/(short)0, C, /*reuse_a=*/false, /*reuse_b=*/false);
  float s = ((D[0] + D[1]) + (D[2] + D[3])) + ((D[4] + D[5]) + (D[6] + D[7]));
  s += __shfl_down(s, 16, 32);
  return s;
}

// Stage a small table into LDS with the gfx1250 async-DMA path.
// Each lane issues one GLOBAL_LOAD_ASYNC_TO_LDS_B32 (GVS mode: saddr = table
// base, vaddr = byte offset, vdst = LDS byte address), then we drain ASYNCcnt.
__device__ __forceinline__ void stage_lds_async(const float* __restrict__ g,
                                                float* s, int n) {
  for (int t = (int)threadIdx.x; t < n; t += (int)blockDim.x) {
    unsigned lds  = (unsigned)(unsigned long long)(&s[t]);   // low 32b = LDS addr
    unsigned voff = (unsigned)t * 4u;
    asm volatile("global_load_async_to_lds_b32 %0, %1, %2"
                 :: "v"(lds), "v"(voff), "s"(g) : "memory");
  }
  asm volatile("s_wait_asynccnt 0x0" ::: "memory");
  __syncthreads();
}

// ---------------- pass 0: zero accumulators ----------------
__global__ void k_zero(float* __restrict__ out, float* __restrict__ gcn,
                       int n_out, int n_gcn) {
  int t = blockIdx.x * blockDim.x + threadIdx.x;
  if (t < n_out) out[t] = 0.0f;   // energy + forces
  if (t < n_gcn) gcn[t] = 0.0f;
}

// ---------------- pass 1: coordination numbers ----------------
__global__ void k_cn(const float* __restrict__ pos, const int* __restrict__ num,
                     const int* __restrict__ nbr, const float* __restrict__ rcov,
                     float* __restrict__ cn_out, int N, int K, int E) {
  __shared__ float s_rcov[128];
  stage_lds_async(rcov, s_rcov, E);
  int t = blockIdx.x * blockDim.x + threadIdx.x;
  int i = t >> 2;
  int sub = t & (SPLIT - 1);
  float cn = 0.0f;
  if (i < N) {
    float pix = pos[3 * i + 0] * CANG2BOHR;
    float piy = pos[3 * i + 1] * CANG2BOHR;
    float piz = pos[3 * i + 2] * CANG2BOHR;
    float rci = s_rcov[num[i]];
    for (int k = sub; k < K; k += SPLIT) {
      int nb = nbr[i * K + k];
      int j = nb < 0 ? 0 : (nb > N - 1 ? N - 1 : nb);
      bool valid = (nb < N) && (nb != i);
      if (!valid) continue;
      float dx = pos[3 * j + 0] * CANG2BOHR - pix;
      float dy = pos[3 * j + 1] * CANG2BOHR - piy;
      float dz = pos[3 * j + 2] * CANG2BOHR - piz;
      float r = sqrtf(dx * dx + dy * dy + dz * dz + 1e-20f);
      float rc = rci + s_rcov[num[j]];
      float s = __expf(-CK1 * (rc * frcp(r) - 1.0f));
      cn += frcp(1.0f + s);
    }
  }
  cn = group_reduce4(cn);
  if (i < N && sub == 0) cn_out[i] = cn;
}

// ---------------- pass 2: per-atom Gaussian weight tables ----------------
// layout (12 floats, 48B stride -> 16B aligned):
//   [0..4]=w[a]  [5..9]=dw/dcn[a]  [10]=sum w  [11]=sum dw/dcn
__global__ void k_weights(const int* __restrict__ num,
                          const float* __restrict__ cnref,
                          const float* __restrict__ cn_in,
                          float* __restrict__ atomW, int N) {
  int i = blockIdx.x * blockDim.x + threadIdx.x;
  if (i >= N) return;
  float cn = cn_in[i];
  int Z = num[i];
  float w[RREF], g[RREF];
  float wsum = 0.0f, sgw = 0.0f;
#pragma unroll
  for (int a = 0; a < RREF; ++a) {
    float d = cn - cnref[Z * RREF + a];
    float e = __expf(CK3 * d * d);
    w[a] = e;
    g[a] = 2.0f * CK3 * d * e;
    wsum += e;
    sgw += g[a];
  }
  float4* dst = (float4*)(atomW + (size_t)i * 12);
  dst[0] = make_float4(w[0], w[1], w[2], w[3]);
  dst[1] = make_float4(w[4], g[0], g[1], g[2]);
  dst[2] = make_float4(g[3], g[4], wsum, sgw);
}

// ---------------- pass 3: energy, dE/dcn, direct dE/dr force ----------------
__global__ void k_pair(const float* __restrict__ pos, const int* __restrict__ num,
                       const int* __restrict__ nbr, const float* __restrict__ r4r2,
                       const float* __restrict__ c6ref,
                       const float* __restrict__ atomW,
                       float* __restrict__ gcn, float* __restrict__ forces,
                       float* __restrict__ energy, int N, int K, int E) {
  __shared__ float s_q[128];
  stage_lds_async(r4r2, s_q, E);
  int t = blockIdx.x * blockDim.x + threadIdx.x;
  int i = t >> 2;
  int sub = t & (SPLIT - 1);
  float e_loc = 0.0f, gcn_i = 0.0f, fx = 0.0f, fy = 0.0f, fz = 0.0f;
  if (i < N) {
    float pix = pos[3 * i + 0] * CANG2BOHR;
    float piy = pos[3 * i + 1] * CANG2BOHR;
    float piz = pos[3 * i + 2] * CANG2BOHR;
    int Zi = num[i];
    float qi = s_q[Zi];
    const float4* Wi4 = (const float4*)(atomW + (size_t)i * 12);
    float4 a0 = Wi4[0], a1 = Wi4[1], a2 = Wi4[2];
    float wi[RREF]  = {a0.x, a0.y, a0.z, a0.w, a1.x};
    float wgi[RREF] = {a1.y, a1.z, a1.w, a2.x, a2.y};
    float wsum_i = a2.z, sgw_i = a2.w;

    for (int k = sub; k < K; k += SPLIT) {
      int nb = nbr[i * K + k];
      int j = nb < 0 ? 0 : (nb > N - 1 ? N - 1 : nb);
      bool valid = (nb < N) && (nb != i);
      if (!valid) continue;
      int Zj = num[j];
      const float* C = c6ref + (size_t)(Zi * E + Zj) * (RREF * RREF);
      __builtin_prefetch(C, 0, 3);
      const float4* Wj4 = (const float4*)(atomW + (size_t)j * 12);
      float4 b0 = Wj4[0], b1 = Wj4[1], b2 = Wj4[2];
      float wj[RREF]  = {b0.x, b0.y, b0.z, b0.w, b1.x};
      float wgj[RREF] = {b1.y, b1.z, b1.w, b2.x, b2.y};
      float wsum_j = b2.z, sgw_j = b2.w;

      float dx = pos[3 * j + 0] * CANG2BOHR - pix;
      float dy = pos[3 * j + 1] * CANG2BOHR - piy;
      float dz = pos[3 * j + 2] * CANG2BOHR - piz;
      float r = sqrtf(dx * dx + dy * dy + dz * dz + 1e-20f);
      float r2 = r * r;

      // u[a] = sum_b C[a][b]*wj[b] ;  tt[b] = sum_a wi[a]*C[a][b]
      float u[RREF], tt[RREF];
#pragma unroll
      for (int a = 0; a < RREF; ++a) { u[a] = 0.0f; tt[a] = 0.0f; }
#pragma unroll
      for (int a = 0; a < RREF; ++a) {
        float wa = wi[a];
#pragma unroll
        for (int b = 0; b < RREF; ++b) {
          float c = C[a * RREF + b];
          u[a] += c * wj[b];
          tt[b] += c * wa;
        }
      }
      float numr = 0.0f, dnum_i = 0.0f, dnum_j = 0.0f;
#pragma unroll
      for (int a = 0; a < RREF; ++a) {
        numr += wi[a] * u[a];
        dnum_i += wgi[a] * u[a];
        dnum_j += wgj[a] * tt[a];
      }
      float invden = frcp(wsum_i * wsum_j + 1e-20f);
      float c6 = numr * invden;

      float qq = 3.0f * qi * s_q[Zj];
      float fd = CA1 * sqrtf(qq) + CA2;
      float f2 = fd * fd;
      float f6 = f2 * f2 * f2;
      float f8 = f6 * f2;
      float r6 = r2 * r2 * r2;
      float r8 = r6 * r2;
      float inv6 = frcp(r6 + f6);
      float inv8 = frcp(r8 + f8);

      float dEdc6 = CS6 * inv6 + CS8 * qq * inv8;        // d e_pair / d c6
      e_loc += c6 * dEdc6;                               // e_pair itself

      float dc6_i = (dnum_i - c6 * (sgw_i * wsum_j)) * invden;
      float dc6_j = (dnum_j - c6 * (wsum_i * sgw_j)) * invden;
      gcn_i += CPREF * dEdc6 * dc6_i;
      atomic_add_f32(&gcn[j], CPREF * dEdc6 * dc6_j);

      float r5 = r2 * r2 * r;
      float r7 = r6 * r;
      float dEdr = -c6 * (6.0f * CS6 * r5 * inv6 * inv6 +
                          8.0f * CS8 * qq * r7 * inv8 * inv8);
      // forces_i += ANG2BOHR*PREF*dEdr*rij/r ; forces_j -= same
      float coef = (CANG2BOHR * CPREF) * dEdr * frcp(r);
      fx += coef * dx;  fy += coef * dy;  fz += coef * dz;
      atomic_add_f32(&forces[3 * j + 0], -coef * dx);
      atomic_add_f32(&forces[3 * j + 1], -coef * dy);
      atomic_add_f32(&forces[3 * j + 2], -coef * dz);
    }
  }
  // combine the SPLIT lanes of each atom (deterministic, one atomic per value on i-side)
  gcn_i = group_reduce4(gcn_i);
  fx = group_reduce4(fx);
  fy = group_reduce4(fy);
  fz = group_reduce4(fz);
  if (i < N && sub == 0) {
    atomic_add_f32(&gcn[i], gcn_i);
    atomic_add_f32(&forces[3 * i + 0], fx);
    atomic_add_f32(&forces[3 * i + 1], fy);
    atomic_add_f32(&forces[3 * i + 2], fz);
  }
  // wave32 energy total via v_wmma_f32_16x16x4_f32, one atomic per wave
  float v = wave_sum_wmma(e_loc * CPREF);
  if ((threadIdx.x & 31u) == 0u) atomic_add_f32(energy, v);
}

// ---------------- pass 4: CN-chain forces (needs complete gcn) ----------------
__global__ void k_cnforce(const float* __restrict__ pos, const int* __restrict__ num,
                          const int* __restrict__ nbr, const float* __restrict__ rcov,
                          const float* __restrict__ gcn, float* __restrict__ forces,
                          int N, int K, int E) {
  __shared__ float s_rcov[128];
  stage_lds_async(rcov, s_rcov, E);
  int t = blockIdx.x * blockDim.x + threadIdx.x;
  int i = t >> 2;
  int sub = t & (SPLIT - 1);
  float fx = 0.0f, fy = 0.0f, fz = 0.0f;
  if (i < N) {
    float pix = pos[3 * i + 0] * CANG2BOHR;
    float piy = pos[3 * i + 1] * CANG2BOHR;
    float piz = pos[3 * i + 2] * CANG2BOHR;
    float rci = s_rcov[num[i]];
    float gi = gcn[i];          // dE/dcn_i (already includes -0.5*H2EV)
    for (int k = sub; k < K; k += SPLIT) {
      int nb = nbr[i * K + k];
      int j = nb < 0 ? 0 : (nb > N - 1 ? N - 1 : nb);
      bool valid = (nb < N) && (nb != i);
      if (!valid) continue;
      float dx = pos[3 * j + 0] * CANG2BOHR - pix;
      float dy = pos[3 * j + 1] * CANG2BOHR - piy;
      float dz = pos[3 * j + 2] * CANG2BOHR - piz;
      float r = sqrtf(dx * dx + dy * dy + dz * dz + 1e-20f);
      float invr = frcp(r);
      float rc = rci + s_rcov[num[j]];
      float s = __expf(-CK1 * (rc * invr - 1.0f));
      float p = frcp(1.0f + s);
      float dcn_dr = -(CK1 * rc * invr * invr) * s * p * p;
      // forces_i += ANG2BOHR * gi * dcn_dr * rij/r ; forces_j -= same
      float coef = CANG2BOHR * gi * dcn_dr * invr;
      fx += coef * dx;  fy += coef * dy;  fz += coef * dz;
      atomic_add_f32(&forces[3 * j + 0], -coef * dx);
      atomic_add_f32(&forces[3 * j + 1], -coef * dy);
      atomic_add_f32(&forces[3 * j + 2], -coef * dz);
    }
  }
  fx = group_reduce4(fx);
  fy = group_reduce4(fy);
  fz = group_reduce4(fz);
  if (i < N && sub == 0) {
    atomic_add_f32(&forces[3 * i + 0], fx);
    atomic_add_f32(&forces[3 * i + 1], fy);
    atomic_add_f32(&forces[3 * i + 2], fz);
  }
}

// ---------------- launch ----------------
extern "C" void kernel_launch(void* const* d_in, const int* in_sizes, int n_in,
                              void* d_out, int out_size, void* d_ws, size_t ws_size,
                              hipStream_t stream) {
  const float* positions = (const float*)d_in[0];
  const int*   numbers   = (const int*)d_in[1];
  const int*   nbr       = (const int*)d_in[2];
  const float* rcov      = (const float*)d_in[3];
  const float* r4r2      = (const float*)d_in[4];
  const float* c6ref     = (const float*)d_in[5];
  const float* cnref     = (const float*)d_in[6];

  const int N = in_sizes[0] / 3;
  const int K = in_sizes[2] / N;
  const int E = in_sizes[3];

  float* out    = (float*)d_out;
  float* energy = out;                       // [0]
  float* forces = out + 1;                   // [1 .. 3N]
  float* cn     = out + 1 + (size_t)3 * N;   // [1+3N .. 1+4N)

  float* atomW = (float*)d_ws;               // N*12 floats
  float* gcn   = atomW + (size_t)N * 12;     // N floats

  const int BS = 256;
  const int ablk = (N + BS - 1) / BS;
  const int sblk = (N * SPLIT + BS - 1) / BS;   // split-over-lanes kernels
  const int zn = 1 + 3 * N;
  const int zmax = zn > N ? zn : N;
  const int zblk = (zmax + BS - 1) / BS;

  k_zero<<<zblk, BS, 0, stream>>>(out, gcn, zn, N);
  k_cn<<<sblk, BS, 0, stream>>>(positions, numbers, nbr, rcov, cn, N, K, E);
  k_weights<<<ablk, BS, 0, stream>>>(numbers, cnref, cn, atomW, N);
  k_pair<<<sblk, BS, 0, stream>>>(positions, numbers, nbr, r4r2, c6ref,
                                  atomW, gcn, forces, energy, N, K, E);
  k_cnforce<<<sblk, BS, 0, stream>>>(positions, numbers, nbr, rcov, gcn,
                                     forces, N, K, E);
}